// LoRA_MoElayer_53987738911386
// MI455X (gfx1250) — compile-verified
//
#include <hip/hip_runtime.h>
#include <math.h>

typedef __attribute__((ext_vector_type(2))) float v2f;
typedef __attribute__((ext_vector_type(4))) float v4f;
typedef __attribute__((ext_vector_type(8))) float v8f;

#define B_SZ   2048
#define N_TOK  196
#define C_DIM  768
#define E_NUM  7
#define R_MAX  128
#define EPS_COMBINE 2.2204460492503131e-16f

// ---------------------------------------------------------------------------
// K1: mean pool over tokens. One block per sample, 192 threads x float4 = 768.
// x is a 1.23 GB single-pass stream -> non-temporal loads keep it out of L2
// (L2 should retain xm + expert weights for the later kernels).
// Also zeroes the expert bucket counters (block 0) before the gating kernel.
// ---------------------------------------------------------------------------
__global__ __launch_bounds__(192) void mean_pool_kernel(
    const float* __restrict__ x, float* __restrict__ xm, int* __restrict__ counts) {
  const int b = blockIdx.x;
  const int t = threadIdx.x;  // 0..191
  if (b == 0 && t < E_NUM) counts[t] = 0;

  const v4f* xb = (const v4f*)(x + (size_t)b * N_TOK * C_DIM);
  v4f acc = {0.f, 0.f, 0.f, 0.f};
#pragma unroll 4
  for (int n = 0; n < N_TOK; ++n) {
    v4f v = __builtin_nontemporal_load(xb + n * (C_DIM / 4) + t);
    acc += v;
  }
  acc *= (1.0f / (float)N_TOK);
  *((v4f*)(xm + (size_t)b * C_DIM) + t) = acc;  // regular store: xm is reused
}

// ---------------------------------------------------------------------------
// K2: hard gating (argmin of squared distance to centers) + bucket-by-expert.
// One wave32 per sample; butterfly reduction; lane 0 appends to bucket list.
// ---------------------------------------------------------------------------
__global__ __launch_bounds__(256) void gate_kernel(
    const float* __restrict__ xm, const float* __restrict__ centers,
    int* __restrict__ counts, int* __restrict__ bucket) {
  const int wave = threadIdx.x >> 5;
  const int lane = threadIdx.x & 31;
  const int b = blockIdx.x * 8 + wave;

  float xv[C_DIM / 32];
  const float* xr = xm + (size_t)b * C_DIM;
#pragma unroll
  for (int i = 0; i < C_DIM / 32; ++i) xv[i] = xr[lane + 32 * i];

  float best = 3.4e38f;
  int bi = 0;
  for (int e = 0; e < E_NUM; ++e) {
    const float* cr = centers + (size_t)e * C_DIM;
    float acc = 0.f;
#pragma unroll
    for (int i = 0; i < C_DIM / 32; ++i) {
      float d = xv[i] - cr[lane + 32 * i];
      acc = fmaf(d, d, acc);
    }
#pragma unroll
    for (int off = 16; off > 0; off >>= 1) acc += __shfl_xor(acc, off, 32);
    if (acc < best) { best = acc; bi = e; }  // uniform across the wave
  }
  if (lane == 0) {
    int pos = atomicAdd(&counts[bi], 1);
    bucket[bi * B_SZ + pos] = b;
  }
}

// ---------------------------------------------------------------------------
// K3: load-balancing loss = (2*cv^2)^2 with unbiased variance over 7 loads.
// ---------------------------------------------------------------------------
__global__ void loss_kernel(const int* __restrict__ counts, float* __restrict__ out_loss) {
  if (threadIdx.x == 0 && blockIdx.x == 0) {
    float s = 0.f;
    for (int e = 0; e < E_NUM; ++e) s += (float)counts[e];
    float mean = s / (float)E_NUM;
    float var = 0.f;
    for (int e = 0; e < E_NUM; ++e) {
      float d = (float)counts[e] - mean;
      var += d * d;
    }
    var /= (float)(E_NUM - 1);
    float cv2 = var / (mean * mean + 1e-10f);
    float l = cv2 + cv2;
    out_loss[0] = l * l;
  }
}

// ---------------------------------------------------------------------------
// K4: h[b, 0:128] = xm[b, :] @ A_e^T for the sample's selected expert, tiled
// 16 samples x 16 r per wave via V_WMMA_F32_16X16X4_F32, K-loop over C=768.
// grid = (B/16 tiles, 7 experts), 8 waves/block cover all 128 r.
// ---------------------------------------------------------------------------
__global__ __launch_bounds__(256) void lora_h_kernel(
    const float* __restrict__ xm, const float* __restrict__ Aw,
    const int* __restrict__ counts, const int* __restrict__ bucket,
    float* __restrict__ h) {
  const int e = blockIdx.y;
  const int cnt = counts[e];
  const int tile0 = blockIdx.x * 16;
  if (tile0 >= cnt) return;  // block-uniform exit, EXEC stays all-ones for WMMA

  __shared__ int sidx[16];
  if (threadIdx.x < 16) {
    int gi = tile0 + (int)threadIdx.x;
    sidx[threadIdx.x] = bucket[e * B_SZ + min(gi, cnt - 1)];
  }
  __syncthreads();

  const int wave = threadIdx.x >> 5;
  const int lane = threadIdx.x & 31;
  const int r0 = wave * 16;          // this wave's 16 output columns (r)
  const int mn = lane & 15;          // A row (sample) == B column (r) index
  const int kk = (lane >> 4) * 2;    // lanes 0-15: K0,K1 ; lanes 16-31: K2,K3

  const float* xrow = xm + (size_t)sidx[mn] * C_DIM + kk;
  const float* wrow = Aw + ((size_t)e * R_MAX + (size_t)(r0 + mn)) * C_DIM + kk;

  v8f acc = {};
  for (int k0 = 0; k0 < C_DIM; k0 += 4) {
    v2f a  = *(const v2f*)(xrow + k0);
    v2f bf = *(const v2f*)(wrow + k0);
    acc = __builtin_amdgcn_wmma_f32_16x16x4_f32(false, a, false, bf,
                                                (short)0, acc, false, false);
  }

  // D layout: VGPR v -> (M=v, N=lane) lanes 0-15 ; (M=v+8, N=lane-16) lanes 16-31
  const int mbase = (lane >> 4) * 8;
#pragma unroll
  for (int v = 0; v < 8; ++v) {
    int mm = v + mbase;
    if (tile0 + mm < cnt)
      h[(size_t)sidx[mm] * R_MAX + r0 + mn] = acc[v];
  }
}

// ---------------------------------------------------------------------------
// K5: out[b, c] = h[b, :] @ Bw_e^T, then one-hot combine: y = log(clamp(
// exp(out), eps)). Same tiling; each wave covers 6 c-tiles (8*6*16 = 768).
// y is write-once -> non-temporal stores.
// ---------------------------------------------------------------------------
__global__ __launch_bounds__(256) void lora_out_kernel(
    const float* __restrict__ h, const float* __restrict__ Bw,
    const int* __restrict__ counts, const int* __restrict__ bucket,
    float* __restrict__ y) {
  const int e = blockIdx.y;
  const int cnt = counts[e];
  const int tile0 = blockIdx.x * 16;
  if (tile0 >= cnt) return;

  __shared__ int sidx[16];
  if (threadIdx.x < 16) {
    int gi = tile0 + (int)threadIdx.x;
    sidx[threadIdx.x] = bucket[e * B_SZ + min(gi, cnt - 1)];
  }
  __syncthreads();

  const int wave = threadIdx.x >> 5;
  const int lane = threadIdx.x & 31;
  const int mn = lane & 15;
  const int kk = (lane >> 4) * 2;
  const float* hrow = h + (size_t)sidx[mn] * R_MAX + kk;
  const int mbase = (lane >> 4) * 8;

  for (int j = 0; j < 6; ++j) {
    const int c0 = (wave * 6 + j) * 16;
    const float* wrow = Bw + ((size_t)e * C_DIM + (size_t)(c0 + mn)) * R_MAX + kk;

    v8f acc = {};
#pragma unroll
    for (int k0 = 0; k0 < R_MAX; k0 += 4) {
      v2f a  = *(const v2f*)(hrow + k0);
      v2f bf = *(const v2f*)(wrow + k0);
      acc = __builtin_amdgcn_wmma_f32_16x16x4_f32(false, a, false, bf,
                                                  (short)0, acc, false, false);
    }

#pragma unroll
    for (int v = 0; v < 8; ++v) {
      int mm = v + mbase;
      if (tile0 + mm < cnt) {
        float o  = acc[v];
        float ev = __expf(o);
        ev = (ev == 0.0f) ? EPS_COMBINE : ev;     // SparseDispatcher eps clamp
        __builtin_nontemporal_store(__logf(ev),
                                    y + (size_t)sidx[mm] * C_DIM + c0 + mn);
      }
    }
  }
}

// ---------------------------------------------------------------------------
extern "C" void kernel_launch(void* const* d_in, const int* in_sizes, int n_in,
                              void* d_out, int out_size, void* d_ws, size_t ws_size,
                              hipStream_t stream) {
  (void)in_sizes; (void)n_in; (void)out_size; (void)ws_size;
  const float* x       = (const float*)d_in[0];   // [B, N, C]
  const float* centers = (const float*)d_in[1];   // [E, C]
  const float* Aw      = (const float*)d_in[2];   // [E, R, C]
  const float* Bw      = (const float*)d_in[3];   // [E, C, R]

  float* y    = (float*)d_out;                    // [B, C]
  float* loss = y + (size_t)B_SZ * C_DIM;         // scalar, appended

  // workspace carving: xm | h | counts | bucket
  float* xm     = (float*)d_ws;                         // B*C floats
  float* h      = xm + (size_t)B_SZ * C_DIM;            // B*R floats
  int*   counts = (int*)(h + (size_t)B_SZ * R_MAX);     // E ints (+pad)
  int*   bucket = counts + 8;                           // E*B ints

  mean_pool_kernel<<<B_SZ, 192, 0, stream>>>(x, xm, counts);
  gate_kernel<<<B_SZ / 8, 256, 0, stream>>>(xm, centers, counts, bucket);
  loss_kernel<<<1, 32, 0, stream>>>(counts, loss);

  dim3 gtiles(B_SZ / 16, E_NUM);
  lora_h_kernel<<<gtiles, 256, 0, stream>>>(xm, Aw, counts, bucket, h);
  lora_out_kernel<<<gtiles, 256, 0, stream>>>(h, Bw, counts, bucket, y);
}